// AdaptiveVectorQuantizer_74268574482507
// MI455X (gfx1250) — compile-verified
//
#include <hip/hip_runtime.h>

typedef __attribute__((ext_vector_type(2))) float v2f;
typedef __attribute__((ext_vector_type(8))) float v8f;

constexpr int D          = 64;
constexpr int P          = 16;
constexpr int LEVELS     = 4;
constexpr int HW         = 4096;      // 64*64
constexpr int C_STRIDE   = 4096;      // channel stride in elements
constexpr int IMG_STRIDE = 262144;    // 64 channels * 4096
constexpr int M_TOTAL    = 131072;    // 32 images * 4096 positions
constexpr int LEVEL_STRIDE = 8388608; // 32 * 262144
constexpr int TILE       = 32;
constexpr int NTILES     = M_TOTAL / TILE; // 4096
constexpr int CBS        = 65;        // padded LDS row stride (bank-conflict free)

constexpr int BLOCKS   = 256;
constexpr int THREADS  = 256;
constexpr int WPB      = THREADS / 32;

__global__ __launch_bounds__(THREADS)
void vq_main(const float* __restrict__ in, const float* __restrict__ cb,
             float* __restrict__ out, float* __restrict__ partial)
{
    __shared__ float s_cb[P * CBS];
    __shared__ float s_x2[WPB][TILE];
    __shared__ float s_smin[WPB][LEVELS][TILE];
    __shared__ int   s_idx[WPB][LEVELS][TILE];
    __shared__ float s_loss[WPB][LEVELS];

    const int tid  = threadIdx.x;
    const int lane = tid & 31;
    const int wv   = tid >> 5;

    // stage codebook into padded LDS (16 x 64, row stride 65 floats)
    for (int i = tid; i < P * D; i += THREADS) {
        int p = i >> 6, c = i & 63;
        s_cb[p * CBS + c] = cb[i];
    }
    __syncthreads();

    const int pcode = lane & 15;             // code index this lane represents (N dim)
    const int koff  = (lane < 16) ? 0 : 2;   // K sub-offset per WMMA f32 16x16x4 layout

    // B fragments: B[k][p] = codebook[p][k]; vgpr0=K(koff), vgpr1=K(koff+1)
    v2f bfrag[16];
#pragma unroll
    for (int ki = 0; ki < 16; ++ki) {
        int k0 = 4 * ki + koff;
        bfrag[ki].x = s_cb[pcode * CBS + k0];
        bfrag[ki].y = s_cb[pcode * CBS + k0 + 1];
    }
    // 0.5 * ||code_p||^2 on each lane
    float halfc2 = 0.f;
#pragma unroll
    for (int c = 0; c < D; ++c) { float v = s_cb[pcode * CBS + c]; halfc2 += v * v; }
    halfc2 *= 0.5f;

    float lossAcc[LEVELS] = {0.f, 0.f, 0.f, 0.f};

    const int waveGlobal = blockIdx.x * WPB + wv;
    const int numWaves   = gridDim.x * WPB;

    for (int tile = waveGlobal; tile < NTILES; tile += numWaves) {
        const int mBase = tile * TILE;
        const int nb    = mBase >> 12;       // image index (TILE divides 4096)
        const int hw    = mBase & 4095;
        const float* inImg = in + (size_t)nb * IMG_STRIDE;

#pragma unroll
        for (int hh = 0; hh < 2; ++hh) {     // two 16-position WMMA tiles
            const int myhw = hw + hh * 16 + pcode;   // position for this lane's M-row
            v8f acc = {0.f,0.f,0.f,0.f,0.f,0.f,0.f,0.f};
            float sq = 0.f;
#pragma unroll
            for (int ki = 0; ki < 16; ++ki) {
                int c0 = 4 * ki + koff;
                v2f a;
                a.x = inImg[(size_t)c0 * C_STRIDE + myhw];
                a.y = inImg[(size_t)(c0 + 1) * C_STRIDE + myhw];
                sq += a.x * a.x + a.y * a.y;
                // D = A(16x4 f32) * B(4x16 f32) + C
                acc = __builtin_amdgcn_wmma_f32_16x16x4_f32(
                        false, a, false, bfrag[ki], (short)0, acc, false, false);
            }
            // full ||x_m||^2: this lane's partial covers K%4 in {koff,koff+1}
            float x2 = sq + __shfl_xor(sq, 16, 32);
            if (lane < 16) s_x2[wv][hh * 16 + lane] = x2;

            // per D-row argmin over codes; butterfly gives prefix-mins = all 4 levels
#pragma unroll
            for (int r = 0; r < 8; ++r) {
                float s  = halfc2 - acc[r];  // monotonic in distance
                int   bi = pcode;
#pragma unroll
                for (int L = 0; L < LEVELS; ++L) {
                    float s2 = __shfl_xor(s, 1 << L, 32);
                    int   i2 = __shfl_xor(bi, 1 << L, 32);
                    bool take = (s2 < s) || (s2 == s && i2 < bi);
                    s  = take ? s2 : s;
                    bi = take ? i2 : bi;
                    if ((lane & 15) == 0) {  // lanes 0 / 16 hold M=r / M=r+8
                        int pos = hh * 16 + r + ((lane >> 4) << 3);
                        s_smin[wv][L][pos] = s;
                        s_idx[wv][L][pos]  = bi;
                    }
                }
            }
        }

        // write phase: lane <-> tile position; streaming NT stores dominate traffic
        const float x2v = s_x2[wv][lane];
        float* outBase = out + (size_t)nb * IMG_STRIDE + (hw + lane);
#pragma unroll
        for (int L = 0; L < LEVELS; ++L) {
            int qi = s_idx[wv][L][lane];
            lossAcc[L] += 2.f * s_smin[wv][L][lane] + x2v;  // sum_c (q-x)^2
            const float* crow = &s_cb[qi * CBS];
            float* o = outBase + (size_t)L * LEVEL_STRIDE;
#pragma unroll 8
            for (int c = 0; c < D; ++c)
                __builtin_nontemporal_store(crow[c], o + (size_t)c * C_STRIDE);
        }
    }

    // deterministic loss reduction: wave shuffle -> block LDS -> per-block partial
#pragma unroll
    for (int L = 0; L < LEVELS; ++L) {
        float v = lossAcc[L];
        for (int m = 16; m >= 1; m >>= 1) v += __shfl_xor(v, m, 32);
        if (lane == 0) s_loss[wv][L] = v;
    }
    __syncthreads();
    if (tid < LEVELS) {
        float sum = 0.f;
        for (int w = 0; w < WPB; ++w) sum += s_loss[w][tid];
        partial[blockIdx.x * LEVELS + tid] = sum;
    }
}

__global__ void vq_finalize(const float* __restrict__ partial,
                            const float* __restrict__ cb,
                            const float* __restrict__ prev,
                            float* __restrict__ out)
{
    float* losses = out + (size_t)LEVELS * LEVEL_STRIDE;
    float* cbout  = losses + LEVELS;
    const int tid = threadIdx.x;

    for (int i = tid; i < P * D; i += blockDim.x) cbout[i] = cb[i];

    if (tid < LEVELS) {
        const int L = tid;
        double sum = 0.0;                       // fixed-order, deterministic
        for (int b = 0; b < BLOCKS; ++b) sum += (double)partial[b * LEVELS + L];
        float quant = (float)(sum / (double)((long long)M_TOTAL * D));
        float prox = 0.f;
        if (L >= 1) {
            int h = 1 << L;                     // nv // 2
            float s = 0.f;
            for (int i = 0; i < h * D; ++i) { float d = prev[i] - cb[i]; s += d * d; }
            prox = (float)L * 0.33f * (s / (float)(h * D));
        }
        float factor = (L <= 1) ? 1.1f : 1.0f;  // quant + 0.1*align (align==quant)
        losses[L] = factor * quant + prox;
    }
}

extern "C" void kernel_launch(void* const* d_in, const int* in_sizes, int n_in,
                              void* d_out, int out_size, void* d_ws, size_t ws_size,
                              hipStream_t stream) {
    const float* in   = (const float*)d_in[0];
    const float* cb   = (const float*)d_in[1];
    const float* prev = (const float*)d_in[2];
    float* out        = (float*)d_out;
    float* partial    = (float*)d_ws;          // BLOCKS*LEVELS floats

    vq_main<<<BLOCKS, THREADS, 0, stream>>>(in, cb, out, partial);
    vq_finalize<<<1, 256, 0, stream>>>(partial, cb, prev, out);
}